// GLIF3_62345745268736
// MI455X (gfx1250) — compile-verified
//
#include <hip/hip_runtime.h>
#include <stdint.h>

// GLIF3 neuron scan for MI455X (gfx1250).
// One thread per (batch, neuron) state machine; sequential over T.
// I_ext streamed HBM -> LDS with gfx1250 async-copy (ASYNCcnt), depth-8
// circular buffer, 7 loads in flight per lane. Outputs stored non-temporal.

#define DT_STEP 0.002f
#define BLK     128
#define DEPTH   8
#define INFLIGHT 7   // keep < DEPTH so the slot being filled is never the slot being read

typedef __attribute__((address_space(1))) int gas_int;
typedef __attribute__((address_space(3))) int las_int;

#if __has_builtin(__builtin_amdgcn_global_load_async_to_lds_b32)
__device__ __forceinline__ void async_g2l_b32(const void* g, void* l) {
  __builtin_amdgcn_global_load_async_to_lds_b32(
      (gas_int*)(uintptr_t)g,
      (las_int*)(uint32_t)(uintptr_t)l,
      /*imm offset*/0, /*cpol*/0);
}
#else
__device__ __forceinline__ void async_g2l_b32(const void* g, void* l) {
  uint32_t lds_off = (uint32_t)(uintptr_t)l;       // low 32 bits give the LDS address
  uint64_t gaddr   = (uint64_t)(uintptr_t)g;
  // VGLOBAL async load: VDST = LDS-address VGPR, VADDR = 64-bit global addr
  asm volatile("global_load_async_to_lds_b32 %0, %1, off"
               :: "v"(lds_off), "v"(gaddr) : "memory");
}
#endif

// Explicit asm for the wait: "memory" clobber pins the following ds_load
// after the ASYNCcnt drain.
#define WAIT_ASYNC(n) asm volatile("s_wait_asynccnt %0" :: "n"(n) : "memory")

struct GlifConst {
  float kv, rvkv, vrest, vreset, vth, slen;
  float ki0, ki1, ri0, ri1, ai0, ai1;
};

__device__ __forceinline__ void glif_step(float cur, const GlifConst& c,
                                          float& V, float& asc0, float& asc1,
                                          float& rsw, float& S_out) {
  float d   = fmaxf(DT_STEP - rsw, 0.0f);
  rsw       = fmaxf(rsw - DT_STEP, 0.0f);
  float Vp  = V;
  float V_  = Vp - d * c.kv * (Vp - c.vrest) + d * (cur + asc0 + asc1) * c.rvkv;
  float S   = (V_ > c.vth) ? 1.0f : 0.0f;
  float head = d * (1.0f - (c.vth - Vp) / (fmaxf(V_, c.vth) - Vp));
  V = (S == 0.0f) ? V_ : c.vreset;
  float sfd = d - head;                       // spike-free dt
  asc0 = asc0 * (1.0f - sfd * c.ki0);
  asc0 = asc0 + S * (asc0 * c.ri0 + c.ai0);
  asc1 = asc1 * (1.0f - sfd * c.ki1);
  asc1 = asc1 + S * (asc1 * c.ri1 + c.ai1);
  rsw  = rsw + S * c.slen - head;
  S_out = S;
}

__global__ __launch_bounds__(BLK) void glif3_scan_kernel(
    const float* __restrict__ I_ext,
    const float* __restrict__ K_V,     const float* __restrict__ RVKV,
    const float* __restrict__ V_rest,  const float* __restrict__ V_reset,
    const float* __restrict__ V_th,    const float* __restrict__ spike_len,
    const float* __restrict__ K_Ij,    const float* __restrict__ R_Ij,
    const float* __restrict__ A_Ij,
    float* __restrict__ outS, float* __restrict__ outV,
    int T, int N) {
  __shared__ float buf[DEPTH * BLK];

  const int tid = threadIdx.x;
  const int n   = blockIdx.x * BLK + tid;
  const int b   = blockIdx.y;
  if (n >= N) return;

  GlifConst c;
  c.kv    = K_V[n];      c.rvkv   = RVKV[n];
  c.vrest = V_rest[n];   c.vreset = V_reset[n];
  c.vth   = V_th[n];     c.slen   = spike_len[n];
  c.ki0   = K_Ij[n];     c.ki1    = K_Ij[N + n];
  c.ri0   = R_Ij[n];     c.ri1    = R_Ij[N + n];
  c.ai0   = A_Ij[n];     c.ai1    = A_Ij[N + n];

  const size_t base = (size_t)b * (size_t)T * (size_t)N + (size_t)n;
  const float* gin = I_ext + base;
  float* gS = outS + base;
  float* gV = outV + base;

  // ---- prologue: fill the async pipeline (INFLIGHT outstanding) ----
  const int pro = (T < INFLIGHT) ? T : INFLIGHT;
  for (int t = 0; t < pro; ++t)
    async_g2l_b32(gin + (size_t)t * N, &buf[(t % DEPTH) * BLK + tid]);

  float V = c.vrest, asc0 = 0.0f, asc1 = 0.0f, rsw = 0.0f, S;

  int t = 0;
  // ---- steady state: one async issue + one consume per step ----
  for (; t + INFLIGHT < T; ++t) {
    WAIT_ASYNC(INFLIGHT - 1);                 // oldest (step t) load landed
    float cur = buf[(t % DEPTH) * BLK + tid]; // ds_load from local slot
    async_g2l_b32(gin + (size_t)(t + INFLIGHT) * N,
                  &buf[((t + INFLIGHT) % DEPTH) * BLK + tid]);
    glif_step(cur, c, V, asc0, asc1, rsw, S);
    __builtin_nontemporal_store(S, gS + (size_t)t * N);
    __builtin_nontemporal_store(V, gV + (size_t)t * N);
  }

  // ---- epilogue: drain remaining INFLIGHT steps ----
  WAIT_ASYNC(0);
  for (; t < T; ++t) {
    float cur = buf[(t % DEPTH) * BLK + tid];
    glif_step(cur, c, V, asc0, asc1, rsw, S);
    __builtin_nontemporal_store(S, gS + (size_t)t * N);
    __builtin_nontemporal_store(V, gV + (size_t)t * N);
  }
}

extern "C" void kernel_launch(void* const* d_in, const int* in_sizes, int n_in,
                              void* d_out, int out_size, void* d_ws, size_t ws_size,
                              hipStream_t stream) {
  const float* I_ext     = (const float*)d_in[0];
  const float* K_V       = (const float*)d_in[1];
  const float* R_V_K_V   = (const float*)d_in[2];
  const float* V_rest    = (const float*)d_in[3];
  const float* V_reset   = (const float*)d_in[4];
  const float* V_thresh  = (const float*)d_in[5];
  const float* spike_len = (const float*)d_in[6];
  const float* K_Ij      = (const float*)d_in[7];
  const float* R_Ij      = (const float*)d_in[8];
  const float* A_Ij      = (const float*)d_in[9];

  const int N  = in_sizes[1];            // K_V has N elements
  const int B  = 16;                     // fixed by setup_inputs
  const int BT = in_sizes[0] / N;        // B*T
  const int T  = BT / B;

  float* outS = (float*)d_out;                       // [B,T,N]
  float* outV = outS + (size_t)B * (size_t)T * N;    // [B,T,N]

  dim3 grid((N + BLK - 1) / BLK, B);
  glif3_scan_kernel<<<grid, dim3(BLK), 0, stream>>>(
      I_ext, K_V, R_V_K_V, V_rest, V_reset, V_thresh, spike_len,
      K_Ij, R_Ij, A_Ij, outS, outV, T, N);
}